// GenerateProposalsOp_3d_23252952941140
// MI455X (gfx1250) — compile-verified
//
#include <hip/hip_runtime.h>
#include <cstdint>

// ---------------------------------------------------------------------------
// GenerateProposalsOp 3D for gfx1250 (MI455X).
// No matmul structure exists in this op, so no WMMA; the CDNA5-specific paths
// exercised are: global_load_async_to_lds_b128 (ASYNCcnt) + s_wait_asynccnt,
// global_prefetch (via __builtin_prefetch), wave32 ballot mask construction,
// LDS-resident IoU tiles.
// Pipeline latency is bound by the inherently serial greedy-NMS chain, so the
// serial kernel keeps all per-iteration state (suppression words, validity)
// in LDS and prefetches the next row's mask words.
// ---------------------------------------------------------------------------

#define N_TOTAL  1572864      // 3 * 32 * 128 * 128
#define SPATIAL  524288       // 32 * 128 * 128
#define PRE_NMS  2000
#define POST_NMS 300
#define NB       2048         // padded candidate count (power of two for sort)

// workspace layout (byte offsets); total = 626688 bytes
#define OFF_HIST1 0           // 2048 u32
#define OFF_HIST2 8192        // 2048 u32
#define OFF_HIST3 16384       // 1024 u32
#define OFF_CTR   20480       // 16 u32: [0]=thr1 [1]=r1 [2]=thr2 [3]=r2
                              //         [4]=thr3 [5]=r3 [6]=countStrict
                              //         [7]=ctrStrict [8]=ctrBoundary
#define OFF_KEYS  24576       // 2048 u32 sortable keys
#define OFF_IDXS  32768       // 2048 u32 flat indices
#define OFF_SCORE 40960       // 2048 f32 sorted scores
#define OFF_ORDER 49152       // 2048 u32 sorted flat indices
#define OFF_VALID 57344       // 2048 u32
#define OFF_BOXES 65536       // SoA: x1,y1,z1,x2,y2,z2 each 2048 f32 (49152 B)
#define OFF_MASKS 114688      // 2000 rows * 64 u32 words (512000 B)

// -------------------- helpers --------------------

__device__ __forceinline__ unsigned int fkey(float f) {
  unsigned int b = __float_as_uint(f);
  return (b & 0x80000000u) ? ~b : (b | 0x80000000u);   // monotonic u32 key
}
__device__ __forceinline__ float keyToFloat(unsigned int k) {
  unsigned int b = (k & 0x80000000u) ? (k & 0x7FFFFFFFu) : ~k;
  return __uint_as_float(b);
}

// scores_flat[t] with t = ((s*H+h)*W+w)*3 + a ; input layout (3,S,H,W)
__device__ __forceinline__ float load_score(const float* __restrict__ cls, int t) {
  int a = t % 3;
  int k = t / 3;
  return cls[a * SPATIAL + k];
}

#if defined(__has_builtin)
#if __has_builtin(__builtin_amdgcn_ballot_w32)
#define BALLOT32(p) __builtin_amdgcn_ballot_w32(p)
#else
#define BALLOT32(p) ((unsigned int)__ballot(p))
#endif
#else
#define BALLOT32(p) ((unsigned int)__ballot(p))
#endif

// CDNA5 async global->LDS copy (ASYNCcnt path). VDST = per-lane LDS byte
// address, VADDR = 64-bit global address, saddr = off.
__device__ __forceinline__ void async_copy_b128(unsigned int lds_addr,
                                                const void* gaddr) {
  asm volatile("global_load_async_to_lds_b128 %0, %1, off"
               :
               : "v"(lds_addr), "v"((unsigned long long)(uintptr_t)gaddr)
               : "memory");
}
__device__ __forceinline__ void wait_async_zero() {
  asm volatile("s_wait_asynccnt 0" ::: "memory");
}

// -------------------- kernels --------------------

__global__ void k_zero(unsigned int* __restrict__ p, int n) {
  int t = blockIdx.x * blockDim.x + threadIdx.x;
  if (t < n) p[t] = 0u;
}

// level-1 histogram over top 11 key bits, LDS-privatized
__global__ __launch_bounds__(256) void k_hist_l0(const float* __restrict__ cls,
                                                 unsigned int* __restrict__ hist1) {
  __shared__ unsigned int lh[2048];
  for (int i = threadIdx.x; i < 2048; i += 256) lh[i] = 0u;
  __syncthreads();
  int stride = gridDim.x * blockDim.x;
  for (int t = blockIdx.x * blockDim.x + threadIdx.x; t < N_TOTAL; t += stride) {
    unsigned int key = fkey(load_score(cls, t));
    atomicAdd(&lh[key >> 21], 1u);
  }
  __syncthreads();
  for (int i = threadIdx.x; i < 2048; i += 256) {
    unsigned int c = lh[i];
    if (c) atomicAdd(&hist1[i], c);
  }
}

// level-2 / level-3 histograms (few matching elements -> direct atomics)
__global__ void k_hist_l12(const float* __restrict__ cls,
                           const unsigned int* __restrict__ ctr,
                           unsigned int* __restrict__ hist, int level) {
  unsigned int thr1 = ctr[0];
  unsigned int thr2 = ctr[2];
  int stride = gridDim.x * blockDim.x;
  for (int t = blockIdx.x * blockDim.x + threadIdx.x; t < N_TOTAL; t += stride) {
    unsigned int key = fkey(load_score(cls, t));
    if ((key >> 21) != thr1) continue;
    if (level == 1) {
      atomicAdd(&hist[(key >> 10) & 2047u], 1u);
    } else {
      if (((key >> 10) & 2047u) != thr2) continue;
      atomicAdd(&hist[key & 1023u], 1u);
    }
  }
}

// single-thread suffix scan of a histogram to find the k-th boundary bin
__global__ void k_pick(unsigned int* __restrict__ ctr,
                       const unsigned int* __restrict__ hist, int nbins, int mode) {
  if (threadIdx.x != 0 || blockIdx.x != 0) return;
  unsigned int target = (mode == 0) ? (unsigned int)PRE_NMS
                                    : ctr[(mode == 1) ? 1 : 3];
  unsigned int cum = 0, thr = 0, r = target;
  for (int b = nbins - 1; b >= 0; --b) {
    unsigned int c = hist[b];
    if (cum + c >= target) { thr = (unsigned int)b; r = target - cum; break; }
    cum += c;
  }
  ctr[mode * 2]     = thr;
  ctr[mode * 2 + 1] = r;
  if (mode == 2) {                       // finalize for compaction
    ctr[6] = (unsigned int)PRE_NMS - r;  // countStrict
    ctr[7] = 0u;                         // ctrStrict
    ctr[8] = 0u;                         // ctrBoundary
  }
}

__global__ void k_compact(const float* __restrict__ cls,
                          unsigned int* __restrict__ ctr,
                          unsigned int* __restrict__ keys,
                          unsigned int* __restrict__ idxs) {
  unsigned int thr1 = ctr[0], thr2 = ctr[2], thr3 = ctr[4];
  unsigned int rB = ctr[5], cStrict = ctr[6];
  int stride = gridDim.x * blockDim.x;
  for (int t = blockIdx.x * blockDim.x + threadIdx.x; t < N_TOTAL; t += stride) {
    unsigned int key = fkey(load_score(cls, t));
    unsigned int b1 = key >> 21;
    if (b1 < thr1) continue;
    int strict;
    if (b1 > thr1) {
      strict = 1;
    } else {
      unsigned int b2 = (key >> 10) & 2047u;
      if (b2 < thr2) continue;
      if (b2 > thr2) {
        strict = 1;
      } else {
        unsigned int b3 = key & 1023u;
        if (b3 < thr3) continue;
        strict = (b3 > thr3) ? 1 : 0;
      }
    }
    unsigned int slot;
    if (strict) {
      slot = atomicAdd(&ctr[7], 1u);
    } else {
      unsigned int p = atomicAdd(&ctr[8], 1u);
      if (p >= rB) continue;
      slot = cStrict + p;
    }
    keys[slot] = key;
    idxs[slot] = (unsigned int)t;
  }
}

// bitonic sort of 2048 (key,idx) pairs, descending key, ties by smaller idx
__global__ __launch_bounds__(1024) void k_sort(const unsigned int* __restrict__ keys,
                                               const unsigned int* __restrict__ idxs,
                                               float* __restrict__ scoreOut,
                                               unsigned int* __restrict__ orderOut) {
  __shared__ unsigned int sk[NB];
  __shared__ unsigned int si[NB];
  int tid = threadIdx.x;
  for (int i = tid; i < NB; i += 1024) {
    if (i < PRE_NMS) { sk[i] = keys[i]; si[i] = idxs[i]; }
    else             { sk[i] = 0u;      si[i] = 0x7FFFFFFFu; }
  }
  __syncthreads();
  for (unsigned int k = 2; k <= NB; k <<= 1) {
    for (unsigned int j = k >> 1; j > 0; j >>= 1) {
      unsigned int i = ((tid & ~(j - 1)) << 1) | (tid & (j - 1));
      unsigned int p = i | j;
      bool asc = ((i & k) != 0);            // final merge -> descending order
      unsigned int ka = sk[i], kb = sk[p], ia = si[i], ib = si[p];
      bool aBetter = (ka > kb) || (ka == kb && ia < ib);
      bool bBetter = (kb > ka) || (kb == ka && ib < ia);
      bool doswap = asc ? aBetter : bBetter;
      if (doswap) { sk[i] = kb; sk[p] = ka; si[i] = ib; si[p] = ia; }
      __syncthreads();
    }
  }
  for (int r = tid; r < PRE_NMS; r += 1024) {
    scoreOut[r] = keyToFloat(sk[r]);
    orderOut[r] = si[r];
  }
}

// decode anchors + deltas -> clipped boxes (SoA) + validity
__global__ __launch_bounds__(256) void k_decode(const float* __restrict__ bbox,
                                                const float* __restrict__ iminfo,
                                                const float* __restrict__ anchors,
                                                const unsigned int* __restrict__ orderArr,
                                                float* __restrict__ boxes,
                                                unsigned int* __restrict__ validArr) {
  int r = blockIdx.x * 256 + threadIdx.x;
  if (r >= PRE_NMS) return;
  unsigned int idx = orderArr[r];
  int a = (int)(idx % 3u);
  int k = (int)(idx / 3u);
  int w = k & 127, h = (k >> 7) & 127, s = k >> 14;
  float sx = w * 4.0f, sy = h * 4.0f, sz = s * 4.0f;
  const float* anc = anchors + a * 6;
  float ax1 = anc[0] + sx, ay1 = anc[1] + sy, az1 = anc[2] + sz;
  float ax2 = anc[3] + sx, ay2 = anc[4] + sy, az2 = anc[5] + sz;
  float aw = ax2 - ax1 + 1.0f, ah = ay2 - ay1 + 1.0f, ad = az2 - az1 + 1.0f;
  float cx = ax1 + 0.5f * aw, cy = ay1 + 0.5f * ah, cz = az1 + 0.5f * ad;
  int base = a * 6;
  float d0 = bbox[(base + 0) * SPATIAL + k];
  float d1 = bbox[(base + 1) * SPATIAL + k];
  float d2 = bbox[(base + 2) * SPATIAL + k];
  float d3 = bbox[(base + 3) * SPATIAL + k];
  float d4 = bbox[(base + 4) * SPATIAL + k];
  float d5 = bbox[(base + 5) * SPATIAL + k];
  float pcx = d0 * aw + cx, pcy = d1 * ah + cy, pcz = d2 * ad + cz;
  float pw = __expf(d3) * aw, ph = __expf(d4) * ah, pd = __expf(d5) * ad;
  float slices = iminfo[0], height = iminfo[1], width = iminfo[2], scale = iminfo[3];
  float x1 = fminf(fmaxf(pcx - 0.5f * pw, 0.0f), width  - 1.0f);
  float y1 = fminf(fmaxf(pcy - 0.5f * ph, 0.0f), height - 1.0f);
  float z1 = fminf(fmaxf(pcz - 0.5f * pd, 0.0f), slices - 1.0f);
  float x2 = fminf(fmaxf(pcx + 0.5f * pw - 1.0f, 0.0f), width  - 1.0f);
  float y2 = fminf(fmaxf(pcy + 0.5f * ph - 1.0f, 0.0f), height - 1.0f);
  float z2 = fminf(fmaxf(pcz + 0.5f * pd - 1.0f, 0.0f), slices - 1.0f);
  boxes[0 * NB + r] = x1; boxes[1 * NB + r] = y1; boxes[2 * NB + r] = z1;
  boxes[3 * NB + r] = x2; boxes[4 * NB + r] = y2; boxes[5 * NB + r] = z2;
  float ss = x2 - x1 + 1.0f;
  float xc = x1 + ss * 0.5f, yc = y1 + ss * 0.5f, zc = z1 + ss * 0.5f;
  bool valid = (ss >= 8.0f * scale) && (xc < width) && (yc < height) && (zc < slices);
  validArr[r] = valid ? 1u : 0u;
}

// suppression bitmask matrix: row i, word wb -> 32 (iou>0.7 && j>i) predicates.
// Strictly upper-triangular: only words wb >= row>>5 are ever non-zero, so only
// those are computed and stored (k_nms_out reads exactly the same set).
__global__ __launch_bounds__(256) void k_masks(const float* __restrict__ boxes,
                                               unsigned int* __restrict__ masks) {
  __shared__ __align__(16) float sb[6 * NB];   // 48 KB of 320 KB/WGP
  __shared__ float svol[NB];
  // Async-copy the SoA box arrays into LDS (CDNA5 ASYNCcnt path).
  unsigned int ldsbase = (unsigned int)(uintptr_t)(&sb[0]);
  for (int t = threadIdx.x; t < (6 * NB) / 4; t += 256) {  // 3072 b128 transfers
    async_copy_b128(ldsbase + (unsigned int)t * 16u,
                    (const char*)boxes + (size_t)t * 16u);
  }
  wait_async_zero();
  __syncthreads();
  for (int j = threadIdx.x; j < NB; j += 256) {
    float v = 0.0f;
    if (j < PRE_NMS)
      v = (sb[3 * NB + j] - sb[j] + 1.0f) *
          (sb[4 * NB + j] - sb[NB + j] + 1.0f) *
          (sb[5 * NB + j] - sb[2 * NB + j] + 1.0f);
    svol[j] = v;
  }
  __syncthreads();
  int wave = threadIdx.x >> 5, lane = threadIdx.x & 31;
  int row = blockIdx.x * 8 + wave;
  if (row >= PRE_NMS) return;
  float x1 = sb[row],          y1 = sb[NB + row],     z1 = sb[2 * NB + row];
  float x2 = sb[3 * NB + row], y2 = sb[4 * NB + row], z2 = sb[5 * NB + row];
  float voli = svol[row];
  for (int wb = row >> 5; wb < 64; ++wb) {
    int j = wb * 32 + lane;
    bool pred = false;
    if (j < PRE_NMS && j > row) {
      float iw  = fmaxf(fminf(x2, sb[3 * NB + j]) - fmaxf(x1, sb[j])          + 1.0f, 0.0f);
      float ih  = fmaxf(fminf(y2, sb[4 * NB + j]) - fmaxf(y1, sb[NB + j])     + 1.0f, 0.0f);
      float idp = fmaxf(fminf(z2, sb[5 * NB + j]) - fmaxf(z1, sb[2 * NB + j]) + 1.0f, 0.0f);
      float inter = iw * ih * idp;
      float iou = inter / (voli + svol[j] - inter);
      pred = iou > 0.7f;
    }
    unsigned int m = BALLOT32(pred);
    if (lane == 0) masks[row * 64 + wb] = m;
  }
}

// single-wave sequential greedy NMS + output emission.
// The serial chain touches only LDS (rem words + async-preloaded validity);
// mask rows are prefetched one iteration ahead to hide global latency.
__global__ __launch_bounds__(32) void k_nms_out(const float* __restrict__ boxes,
                                                const unsigned int* __restrict__ validArr,
                                                const unsigned int* __restrict__ masks,
                                                const float* __restrict__ scoreArr,
                                                const unsigned int* __restrict__ orderArr,
                                                float* __restrict__ out) {
  __shared__ unsigned int rem[64];
  __shared__ __align__(16) unsigned int svalid[NB];   // 8 KB
  __shared__ int kept[POST_NMS];
  int lane = threadIdx.x;
  // async-preload the validity flags so the sequential loop never hits global
  unsigned int ldsbase = (unsigned int)(uintptr_t)(&svalid[0]);
  for (int t = lane; t < NB / 4; t += 32) {            // 512 b128 transfers
    async_copy_b128(ldsbase + (unsigned int)t * 16u,
                    (const char*)validArr + (size_t)t * 16u);
  }
  rem[lane] = 0u;
  rem[lane + 32] = 0u;
  wait_async_zero();
  __syncthreads();
  int nk = 0;   // uniform across the wave
  for (int i = 0; i < PRE_NMS; ++i) {
    if (i + 1 < PRE_NMS) {   // warm L2/L0 for the next row's mask words
      __builtin_prefetch(&masks[(size_t)(i + 1) * 64 + lane], 0, 1);
      __builtin_prefetch(&masks[(size_t)(i + 1) * 64 + 32 + lane], 0, 1);
    }
    bool keep = (nk < POST_NMS) && (svalid[i] != 0u) &&
                !((rem[i >> 5] >> (i & 31)) & 1u);
    if (keep) {
      if (lane == 0) kept[nk] = i;
      int w0 = i >> 5;       // words below w0 are structurally zero
      if (lane >= w0)      rem[lane]      |= masks[i * 64 + lane];
      if (lane + 32 >= w0) rem[lane + 32] |= masks[i * 64 + 32 + lane];
      ++nk;
    }
    __syncthreads();
  }
  // outputs: rois(300x7) | scores(300) | keep_idx(300) | sel_valid(300)
  for (int r = lane; r < POST_NMS; r += 32) {
    bool v = r < nk;
    float x1 = 0.f, y1 = 0.f, z1 = 0.f, x2 = 0.f, y2 = 0.f, z2 = 0.f;
    float sc = 0.f, ki = -1.f, sv = 0.f;
    if (v) {
      int i = kept[r];
      x1 = boxes[i];          y1 = boxes[NB + i];     z1 = boxes[2 * NB + i];
      x2 = boxes[3 * NB + i]; y2 = boxes[4 * NB + i]; z2 = boxes[5 * NB + i];
      sc = scoreArr[i];
      ki = (float)(int)orderArr[i];
      sv = 1.0f;
    }
    out[r * 7 + 0] = 0.0f;
    out[r * 7 + 1] = x1; out[r * 7 + 2] = y1; out[r * 7 + 3] = z1;
    out[r * 7 + 4] = x2; out[r * 7 + 5] = y2; out[r * 7 + 6] = z2;
    out[POST_NMS * 7 + r] = sc;
    out[POST_NMS * 8 + r] = ki;
    out[POST_NMS * 9 + r] = sv;
  }
}

// -------------------- launch --------------------

extern "C" void kernel_launch(void* const* d_in, const int* in_sizes, int n_in,
                              void* d_out, int out_size, void* d_ws, size_t ws_size,
                              hipStream_t stream) {
  (void)in_sizes; (void)n_in; (void)out_size; (void)ws_size;
  const float* cls     = (const float*)d_in[0];  // (1,3,32,128,128)
  const float* bbox    = (const float*)d_in[1];  // (1,18,32,128,128)
  const float* iminfo  = (const float*)d_in[2];  // (1,4)
  const float* anchors = (const float*)d_in[3];  // (3,6)
  float* out = (float*)d_out;

  unsigned char* ws = (unsigned char*)d_ws;
  unsigned int* hist1    = (unsigned int*)(ws + OFF_HIST1);
  unsigned int* hist2    = (unsigned int*)(ws + OFF_HIST2);
  unsigned int* hist3    = (unsigned int*)(ws + OFF_HIST3);
  unsigned int* ctr      = (unsigned int*)(ws + OFF_CTR);
  unsigned int* keys     = (unsigned int*)(ws + OFF_KEYS);
  unsigned int* idxs     = (unsigned int*)(ws + OFF_IDXS);
  float*        scoreArr = (float*)       (ws + OFF_SCORE);
  unsigned int* orderArr = (unsigned int*)(ws + OFF_ORDER);
  unsigned int* validArr = (unsigned int*)(ws + OFF_VALID);
  float*        boxes    = (float*)       (ws + OFF_BOXES);
  unsigned int* masks    = (unsigned int*)(ws + OFF_MASKS);

  // zero hist1..3, ctr, keys, idxs (everything before OFF_SCORE) = 40960 B
  int nzero = 40960 / 4;
  k_zero<<<(nzero + 255) / 256, 256, 0, stream>>>((unsigned int*)ws, nzero);

  // 3-level radix select of top PRE_NMS scores
  k_hist_l0 <<<1024, 256, 0, stream>>>(cls, hist1);
  k_pick    <<<1, 1, 0, stream>>>(ctr, hist1, 2048, 0);
  k_hist_l12<<<512, 256, 0, stream>>>(cls, ctr, hist2, 1);
  k_pick    <<<1, 1, 0, stream>>>(ctr, hist2, 2048, 1);
  k_hist_l12<<<512, 256, 0, stream>>>(cls, ctr, hist3, 2);
  k_pick    <<<1, 1, 0, stream>>>(ctr, hist3, 1024, 2);
  k_compact <<<1024, 256, 0, stream>>>(cls, ctr, keys, idxs);

  // sort candidates descending by score (ties by index)
  k_sort<<<1, 1024, 0, stream>>>(keys, idxs, scoreArr, orderArr);

  // decode boxes + validity
  k_decode<<<(PRE_NMS + 255) / 256, 256, 0, stream>>>(bbox, iminfo, anchors,
                                                      orderArr, boxes, validArr);

  // suppression bitmask matrix (LDS tile loaded via async-to-LDS)
  k_masks<<<PRE_NMS / 8, 256, 0, stream>>>(boxes, masks);

  // sequential greedy NMS + emit all 3000 outputs
  k_nms_out<<<1, 32, 0, stream>>>(boxes, validArr, masks, scoreArr, orderArr, out);
}